// RoiAlignLayer_19834158973716
// MI455X (gfx1250) — compile-verified
//
#include <hip/hip_runtime.h>
#include <stdint.h>

// ---------------------------------------------------------------------------
// FPN ROI-Align for MI455X (gfx1250).
// Memory-bound gather (~0.5 GB traffic -> ~21us @ 23.3 TB/s; pyramid fits in
// 192MB L2). Uses CDNA5 async global->LDS copies (ASYNCcnt) with double
// buffering to hide gather latency; no WMMA (op is not a dense matmul).
// ---------------------------------------------------------------------------

#define POOLN    7
#define GRID_PTS 49
#define CHANNELS 256

#ifndef __has_builtin
#define __has_builtin(x) 0
#endif

#if __has_builtin(__builtin_amdgcn_global_load_async_to_lds_b128)
#define HAVE_ASYNC_LD 1
#endif
#if __has_builtin(__builtin_amdgcn_s_wait_asynccnt)
#define HAVE_ASYNC_WAIT 1
#endif

typedef int v4i __attribute__((vector_size(16)));
typedef __attribute__((address_space(1))) v4i g_v4i;   // global int4
typedef __attribute__((address_space(3))) v4i l_v4i;   // LDS int4
typedef __attribute__((address_space(3))) void lds_void_t;

// Async copy of 16 bytes (per lane) global -> LDS. Tracked by ASYNCcnt.
__device__ __forceinline__ void async_copy_b128(float* lds_dst, const float* gsrc) {
#if defined(HAVE_ASYNC_LD)
  __builtin_amdgcn_global_load_async_to_lds_b128(
      (g_v4i*)gsrc, (l_v4i*)lds_dst, 0, 0);
#else
  unsigned loff = (unsigned)(uintptr_t)(lds_void_t*)lds_dst;
  asm volatile("global_load_async_to_lds_b128 %0, %1, off"
               :: "v"(loff), "v"(gsrc) : "memory");
#endif
}

template <int N>
__device__ __forceinline__ void wait_asynccnt() {
#if defined(HAVE_ASYNC_WAIT)
  __builtin_amdgcn_s_wait_asynccnt(N);
#else
  asm volatile("s_wait_asynccnt %0" :: "n"(N));
#endif
  asm volatile("" ::: "memory");  // keep LDS reads after the wait
}

__device__ __forceinline__ int iclamp(int v, int lo, int hi) {
  return v < lo ? lo : (v > hi ? hi : v);
}

// ---------------------------------------------------------------------------
// Kernel 1: per-ROI level + stable counting-sort positions (matches stable
// jnp.argsort(roi_level)). Single block, 256 threads; R <= 2048.
// ---------------------------------------------------------------------------
__global__ __launch_bounds__(256)
void roi_level_pos_kernel(const float* __restrict__ rois,
                          const int* __restrict__ image_h,
                          const int* __restrict__ image_w,
                          int R,
                          int* __restrict__ pos_out,
                          int* __restrict__ lvl_out) {
  __shared__ int lvl_s[2048];
  __shared__ int chunkCnt[256][4];
  __shared__ int chunkBase[256][4];
  __shared__ int levelBase[4];

  const int t = threadIdx.x;
  const float area  = (float)(*image_h) * (float)(*image_w);
  const float denom = 224.0f / sqrtf(area);   // CANONICAL / sqrt(area)

  const int per = (R + 255) / 256;
  const int beg = t * per;
  const int end = (beg + per < R) ? beg + per : R;

  int cnt[4] = {0, 0, 0, 0};
  for (int r = beg; r < end; ++r) {
    float y1 = rois[4 * r + 0], x1 = rois[4 * r + 1];
    float y2 = rois[4 * r + 2], x2 = rois[4 * r + 3];
    float h = y2 - y1, w = x2 - x1;
    float hw = fmaxf(h * w, 1e-12f);
    float spec = log2f(sqrtf(hw) / denom);
    int l = 4 + (int)rintf(spec);             // rintf == round-half-even
    l = iclamp(l, 2, 5) - 2;                  // 0..3
    lvl_s[r] = l;
    cnt[l]++;
  }
  chunkCnt[t][0] = cnt[0]; chunkCnt[t][1] = cnt[1];
  chunkCnt[t][2] = cnt[2]; chunkCnt[t][3] = cnt[3];
  __syncthreads();

  if (t < 4) {  // exclusive prefix over chunks, per level (4 threads in parallel)
    int acc = 0;
    for (int i = 0; i < 256; ++i) { chunkBase[i][t] = acc; acc += chunkCnt[i][t]; }
    levelBase[t] = acc;  // total per level (temporarily)
  }
  __syncthreads();
  if (t == 0) {          // exclusive prefix over the 4 level totals
    int acc = 0;
    for (int l = 0; l < 4; ++l) { int tot = levelBase[l]; levelBase[l] = acc; acc += tot; }
  }
  __syncthreads();

  int run[4] = {0, 0, 0, 0};
  for (int r = beg; r < end; ++r) {
    int l = lvl_s[r];
    pos_out[r] = levelBase[l] + chunkBase[t][l] + run[l];
    run[l]++;
    lvl_out[r] = l;
  }
}

// ---------------------------------------------------------------------------
// Kernel 2: pooling. One block per (roi, py) row; 64 threads, each owning 4
// consecutive channels (float4). Double-buffered async global->LDS staging of
// the 4 bilinear corner rows (4 x 1KB per grid point).
// ---------------------------------------------------------------------------
struct PointInfo { int i00, i01, i10, i11; float wy, wx, valid; };

__global__ __launch_bounds__(64)
void roi_pool_kernel(const float* __restrict__ rois,
                     const float* __restrict__ p2, const float* __restrict__ p3,
                     const float* __restrict__ p4, const float* __restrict__ p5,
                     const int* __restrict__ pos, const int* __restrict__ lvl,
                     int N, float* __restrict__ out) {
  const int blk = blockIdx.x;
  const int r   = blk / POOLN;
  const int py  = blk % POOLN;
  const int t   = threadIdx.x;            // 0..63 -> channels 4t..4t+3
  const int b   = r / N;
  const int l   = lvl[r];

  const float* feat; int H;
  if      (l == 0) { feat = p2; H = 256; }
  else if (l == 1) { feat = p3; H = 128; }
  else if (l == 2) { feat = p4; H = 64;  }
  else             { feat = p5; H = 32;  }
  const int W = H;

  const float y1 = rois[4 * r + 0], x1 = rois[4 * r + 1];
  const float y2 = rois[4 * r + 2], x2 = rois[4 * r + 3];
  const float dh = y2 - y1, dw = x2 - x1;

  const float* fbase = feat + (size_t)b * H * W * CHANNELS + 4 * t;
  float* obase = out + ((size_t)pos[r] * GRID_PTS + (size_t)py * POOLN) * CHANNELS + 4 * t;

  __shared__ __align__(16) float buf[2][4 * CHANNELS];

  const float fy   = (float)py / 6.0f;
  const float in_y = (y1 + fy * dh) * (float)(H - 1);
  const float y0f  = floorf(in_y);
  const float wy   = in_y - y0f;
  const int   y0i  = (int)y0f;
  const int   y0c  = iclamp(y0i, 0, H - 1);
  const int   y1c  = iclamp(y0i + 1, 0, H - 1);
  const float vy   = (in_y >= 0.0f && in_y <= (float)(H - 1)) ? 1.0f : 0.0f;

  auto calc = [&](int px) {
    PointInfo q;
    float fx   = (float)px / 6.0f;
    float in_x = (x1 + fx * dw) * (float)(W - 1);
    float x0f  = floorf(in_x);
    q.wx = in_x - x0f;
    q.wy = wy;
    int x0i = (int)x0f;
    int x0c = iclamp(x0i, 0, W - 1);
    int x1c = iclamp(x0i + 1, 0, W - 1);
    q.i00 = (y0c * W + x0c) * CHANNELS;
    q.i01 = (y0c * W + x1c) * CHANNELS;
    q.i10 = (y1c * W + x0c) * CHANNELS;
    q.i11 = (y1c * W + x1c) * CHANNELS;
    q.valid = ((in_x >= 0.0f && in_x <= (float)(W - 1)) ? 1.0f : 0.0f) * vy;
    return q;
  };

  auto issue = [&](int px, int slot) {  // 4 async b128 per lane -> ASYNCcnt += 4
    PointInfo q = calc(px);
    async_copy_b128(&buf[slot][0 * CHANNELS + 4 * t], fbase + q.i00);
    async_copy_b128(&buf[slot][1 * CHANNELS + 4 * t], fbase + q.i01);
    async_copy_b128(&buf[slot][2 * CHANNELS + 4 * t], fbase + q.i10);
    async_copy_b128(&buf[slot][3 * CHANNELS + 4 * t], fbase + q.i11);
  };

  issue(0, 0);
  for (int px = 0; px < POOLN; ++px) {
    const int slot = px & 1;
    if (px + 1 < POOLN) { issue(px + 1, slot ^ 1); wait_asynccnt<4>(); }
    else                { wait_asynccnt<0>(); }

    PointInfo q = calc(px);
    float4 v00 = *(const float4*)&buf[slot][0 * CHANNELS + 4 * t];
    float4 v01 = *(const float4*)&buf[slot][1 * CHANNELS + 4 * t];
    float4 v10 = *(const float4*)&buf[slot][2 * CHANNELS + 4 * t];
    float4 v11 = *(const float4*)&buf[slot][3 * CHANNELS + 4 * t];
    float owx = 1.0f - q.wx, owy = 1.0f - q.wy;
    float4 o;
    o.x = ((v00.x * owx + v01.x * q.wx) * owy + (v10.x * owx + v11.x * q.wx) * q.wy) * q.valid;
    o.y = ((v00.y * owx + v01.y * q.wx) * owy + (v10.y * owx + v11.y * q.wx) * q.wy) * q.valid;
    o.z = ((v00.z * owx + v01.z * q.wx) * owy + (v10.z * owx + v11.z * q.wx) * q.wy) * q.valid;
    o.w = ((v00.w * owx + v01.w * q.wx) * owy + (v10.w * owx + v11.w * q.wx) * q.wy) * q.valid;
    *(float4*)&obase[px * CHANNELS] = o;
  }
}

// ---------------------------------------------------------------------------
extern "C" void kernel_launch(void* const* d_in, const int* in_sizes, int n_in,
                              void* d_out, int out_size, void* d_ws, size_t ws_size,
                              hipStream_t stream) {
  const float* rois    = (const float*)d_in[0];
  const int*   image_h = (const int*)d_in[1];
  const int*   image_w = (const int*)d_in[2];
  const float* p2      = (const float*)d_in[3];
  const float* p3      = (const float*)d_in[4];
  const float* p4      = (const float*)d_in[5];
  const float* p5      = (const float*)d_in[6];

  const int B = in_sizes[3] / (256 * 256 * CHANNELS);   // p2 is B x 256 x 256 x C
  const int R = in_sizes[0] / 4;                        // total ROIs
  const int N = R / B;                                  // ROIs per batch image

  int* pos = (int*)d_ws;
  int* lvl = pos + R;

  roi_level_pos_kernel<<<1, 256, 0, stream>>>(rois, image_h, image_w, R, pos, lvl);
  roi_pool_kernel<<<R * POOLN, 64, 0, stream>>>(rois, p2, p3, p4, p5, pos, lvl,
                                                N, (float*)d_out);
}